// EfficientRNN_74277164417180
// MI455X (gfx1250) — compile-verified
//
#include <hip/hip_runtime.h>
#include <math.h>

// Problem constants
#define BB 64
#define TT 512
#define II 256
#define HH 512
#define LL 3
#define SS 3

typedef float v2f __attribute__((ext_vector_type(2)));
typedef float v8f __attribute__((ext_vector_type(8)));

// ---------------------------------------------------------------------------
// Generic f32 WMMA GEMM with bias:
//   C[s](M x N) = A[s](M x K) * W[s](N x K)^T + bias[s](N)
// One 16x16 output tile per wave (32 threads). Uses V_WMMA_F32_16X16X4_F32
// (exact f32 math -> matches the f32 reference numerics).
// A-frag layout: lane holds M = lane&15, K = kk + 2*(lane>>4) + {0,1}
// B-frag layout: lane holds N = lane&15, same K split (B[k][n] = W[n*K+k])
// C layout: vgpr i -> M = i + 8*(lane>>4), N = lane&15
// ---------------------------------------------------------------------------
__global__ void __launch_bounds__(32)
wmma_gemm_bias(const float* __restrict__ A, long sAs, int lda,
               const float* __restrict__ W, long sWs, int K,
               const float* __restrict__ bias, int sBs,
               float* __restrict__ C, long sCs, int ldc)
{
    int s    = blockIdx.z;
    int n0   = blockIdx.x * 16;
    int m0   = blockIdx.y * 16;
    int lane = threadIdx.x;
    int l15  = lane & 15;
    int lhalf = (lane >> 4) << 1;   // 0 or 2

    const float* Ap = A + (size_t)s * sAs + (size_t)(m0 + l15) * lda + lhalf;
    const float* Wp = W + (size_t)s * sWs + (size_t)(n0 + l15) * K   + lhalf;

    v8f c = {};
#pragma unroll 4
    for (int kk = 0; kk < K; kk += 4) {
        v2f a = *(const v2f*)(Ap + kk);
        v2f b = *(const v2f*)(Wp + kk);
        c = __builtin_amdgcn_wmma_f32_16x16x4_f32(false, a, false, b,
                                                  (short)0, c, false, false);
    }

    float bb  = bias[(size_t)s * sBs + n0 + l15];
    int mtop  = (lane >> 4) << 3;
#pragma unroll
    for (int i = 0; i < 8; i++) {
        C[(size_t)s * sCs + (size_t)(m0 + mtop + i) * ldc + n0 + l15] = c[i] + bb;
    }
}

// ---------------------------------------------------------------------------
// Xc[r, k] = dot(x[r, :], Ws[k, H:H+I])   for r = b*T + t, k < 3.
// WMMA with N padded to 16: lanes with col >= 3 read row 0 and are zeroed.
// ---------------------------------------------------------------------------
__global__ void __launch_bounds__(32)
xc_gemm(const float* __restrict__ x, const float* __restrict__ Ws,
        float* __restrict__ Xc)
{
    const int K = II, LDW = HH + II;
    int m0   = blockIdx.x * 16;
    int lane = threadIdx.x;
    int l15  = lane & 15;
    int lhalf = (lane >> 4) << 1;
    bool valid = (l15 < SS);
    int  wr  = valid ? l15 : 0;
    float vm = valid ? 1.0f : 0.0f;

    const float* Ap = x  + (size_t)(m0 + l15) * II + lhalf;
    const float* Wp = Ws + (size_t)wr * LDW + HH + lhalf;

    v8f c = {};
#pragma unroll 4
    for (int kk = 0; kk < K; kk += 4) {
        v2f a = *(const v2f*)(Ap + kk);
        v2f b = *(const v2f*)(Wp + kk);
        b.x *= vm; b.y *= vm;
        c = __builtin_amdgcn_wmma_f32_16x16x4_f32(false, a, false, b,
                                                  (short)0, c, false, false);
    }

    if (valid) {
        int mtop = (lane >> 4) << 3;
#pragma unroll
        for (int i = 0; i < 8; i++)
            Xc[(size_t)(m0 + mtop + i) * SS + l15] = c[i];
    }
}

// ---------------------------------------------------------------------------
// GRU with zero initial state:
//   r = sig(gi_r + bhh_r); z = sig(gi_z + bhh_z);
//   n = tanh(gi_n + r*bhh_n); h = (1-z)*n
// gi rows are (s*B + b) with ld = 3H; gi already contains x@W^T + bih.
// ---------------------------------------------------------------------------
__global__ void gru_kernel(const float* __restrict__ gi,
                           const float* __restrict__ bhh, int sBs,
                           float* __restrict__ h)
{
    int idx = blockIdx.x * 256 + threadIdx.x;       // < S*B*H = 98304
    int row = idx >> 9;                             // s*B + b
    int hh  = idx & (HH - 1);
    int s   = row >> 6;
    const float* g  = gi  + (size_t)row * (3 * HH);
    const float* bh = bhh + (size_t)s * sBs;
    float r = 1.0f / (1.0f + expf(-(g[hh]          + bh[hh])));
    float z = 1.0f / (1.0f + expf(-(g[HH + hh]     + bh[HH + hh])));
    float n = tanhf(g[2 * HH + hh] + r * bh[2 * HH + hh]);
    h[idx] = (1.0f - z) * n;
}

// wcol[h] = sum_h' Wl[h', h]  (column sums; bl cancels in the softmax)
__global__ void colsum_kernel(const float* __restrict__ Wl, float* __restrict__ wcol)
{
    int h = blockIdx.x * blockDim.x + threadIdx.x;
    if (h < HH) {
        float a = 0.f;
        for (int r = 0; r < HH; r++) a += Wl[(size_t)r * HH + h];
        wcol[h] = a;
    }
}

// ---------------------------------------------------------------------------
// Per (s, b): dot(h_all[s,b,l,:], wcol) -> softmax over l -> sh ->
// A3[s,b,k] = dot(sh, Ws[k,:H]) + bs[k]
// ---------------------------------------------------------------------------
__global__ void __launch_bounds__(256)
attn_sh_kernel(const float* __restrict__ h0, const float* __restrict__ h1,
               const float* __restrict__ h2, const float* __restrict__ wcol,
               const float* __restrict__ Ws, const float* __restrict__ bs,
               float* __restrict__ A3)
{
    __shared__ float red[256];
    __shared__ float attn[3];
    int tid = threadIdx.x;
    int sb  = blockIdx.x;   // s*B + b

    const float* hp0 = h0 + (size_t)sb * HH;
    const float* hp1 = h1 + (size_t)sb * HH;
    const float* hp2 = h2 + (size_t)sb * HH;

    auto blockReduce = [&](float v) -> float {
        red[tid] = v; __syncthreads();
        for (int off = 128; off > 0; off >>= 1) {
            if (tid < off) red[tid] += red[tid + off];
            __syncthreads();
        }
        float r = red[0]; __syncthreads();
        return r;
    };

    float a0 = 0.f, a1 = 0.f, a2 = 0.f;
    for (int h = tid; h < HH; h += 256) {
        float w = wcol[h];
        a0 += hp0[h] * w; a1 += hp1[h] * w; a2 += hp2[h] * w;
    }
    float d0 = blockReduce(a0);
    float d1 = blockReduce(a1);
    float d2 = blockReduce(a2);

    if (tid == 0) {
        float m  = fmaxf(d0, fmaxf(d1, d2));
        float e0 = expf(d0 - m), e1 = expf(d1 - m), e2 = expf(d2 - m);
        float inv = 1.0f / (e0 + e1 + e2);
        attn[0] = e0 * inv; attn[1] = e1 * inv; attn[2] = e2 * inv;
    }
    __syncthreads();

    float p0 = 0.f, p1 = 0.f, p2 = 0.f;
    for (int h = tid; h < HH; h += 256) {
        float sh = attn[0] * hp0[h] + attn[1] * hp1[h] + attn[2] * hp2[h];
        p0 += sh * Ws[0 * (HH + II) + h];
        p1 += sh * Ws[1 * (HH + II) + h];
        p2 += sh * Ws[2 * (HH + II) + h];
    }
    float q0 = blockReduce(p0);
    float q1 = blockReduce(p1);
    float q2 = blockReduce(p2);
    if (tid == 0) {
        A3[(size_t)sb * 3 + 0] = q0 + bs[0];
        A3[(size_t)sb * 3 + 1] = q1 + bs[1];
        A3[(size_t)sb * 3 + 2] = q2 + bs[2];
    }
}

// ---------------------------------------------------------------------------
// Deterministic sequential scan: per t, 64x3 softmax, penalized argmax.
// One block of 64 threads (b = tid). Serial per-k sums keep it deterministic.
// ---------------------------------------------------------------------------
__global__ void __launch_bounds__(64)
scan_kernel(const float* __restrict__ A3, const float* __restrict__ Xc,
            int* __restrict__ cur)
{
    __shared__ float p[3];
    __shared__ int   scur;
    __shared__ float sbuf[3 * 64];
    __shared__ float score[3];
    int tid = threadIdx.x;
    if (tid == 0) { p[0] = 0.5f; p[1] = 1.f; p[2] = 1.f; scur = 0; cur[0] = 0; }
    __syncthreads();

    for (int t = 1; t < TT; t++) {
        int s = scur;
        const float* a  = A3 + (size_t)(s * BB + tid) * 3;
        const float* xc = Xc + ((size_t)tid * TT + t) * 3;
        float l0 = a[0] + xc[0], l1 = a[1] + xc[1], l2 = a[2] + xc[2];
        float m  = fmaxf(l0, fmaxf(l1, l2));
        float e0 = expf(l0 - m), e1 = expf(l1 - m), e2 = expf(l2 - m);
        float inv = 1.0f / (e0 + e1 + e2);
        sbuf[0 * 64 + tid] = e0 * inv;
        sbuf[1 * 64 + tid] = e1 * inv;
        sbuf[2 * 64 + tid] = e2 * inv;
        __syncthreads();
        if (tid < 3) {
            float acc = 0.f;
            for (int b = 0; b < 64; b++) acc += sbuf[tid * 64 + b];
            score[tid] = acc * p[tid];
        }
        __syncthreads();
        if (tid == 0) {
            int am = 0;                               // first-max wins (jnp.argmax)
            if (score[1] > score[am]) am = 1;
            if (score[2] > score[am]) am = 2;
            cur[t] = am;
            p[am] *= 0.5f;
            float mx = fmaxf(p[0], fmaxf(p[1], p[2]));
            p[0] /= mx; p[1] /= mx; p[2] /= mx;
            scur = am;
        }
        __syncthreads();
    }
}

// ---------------------------------------------------------------------------
// outputs[b, t, h] = h2[cur[t]][b][h];  h_fin[b, l, h] = h_l[cur[T-1]][b][h]
// ---------------------------------------------------------------------------
__global__ void write_out(const float* __restrict__ h0, const float* __restrict__ h1,
                          const float* __restrict__ h2, const int* __restrict__ cur,
                          float* __restrict__ out)
{
    size_t idx = (size_t)blockIdx.x * 256 + threadIdx.x;
    const size_t NOUT = (size_t)BB * TT * HH;
    if (idx < NOUT) {
        int    h = idx & (HH - 1);
        size_t r = idx >> 9;             // b*T + t
        int    t = (int)(r & (TT - 1));
        int    b = (int)(r >> 9);
        int    s = cur[t];
        out[idx] = h2[((size_t)(s * BB + b)) * HH + h];
    } else {
        size_t j = idx - NOUT;
        if (j < (size_t)BB * LL * HH) {
            int    h = j & (HH - 1);
            size_t r = j >> 9;           // b*L + l
            int    l = (int)(r % LL);
            int    b = (int)(r / LL);
            int    s = cur[TT - 1];
            const float* hl = (l == 0) ? h0 : ((l == 1) ? h1 : h2);
            out[NOUT + j] = hl[((size_t)(s * BB + b)) * HH + h];
        }
    }
}

// ---------------------------------------------------------------------------
extern "C" void kernel_launch(void* const* d_in, const int* in_sizes, int n_in,
                              void* d_out, int out_size, void* d_ws, size_t ws_size,
                              hipStream_t stream)
{
    const float* x    = (const float*)d_in[0];
    const float* Wl   = (const float*)d_in[1];
    // bl (d_in[2]) cancels in the softmax — unused.
    const float* Ws   = (const float*)d_in[3];
    const float* bs   = (const float*)d_in[4];
    const float* Wih0 = (const float*)d_in[5];
    const float* bih0 = (const float*)d_in[6];
    const float* bhh0 = (const float*)d_in[7];
    const float* WihL = (const float*)d_in[8];
    const float* bihL = (const float*)d_in[9];
    const float* bhhL = (const float*)d_in[10];
    float* out = (float*)d_out;

    // Workspace carve-up (floats): ~2.8 MB total
    float* ws   = (float*)d_ws;
    float* gi   = ws;                         // S*B*3H = 294912
    float* h0   = gi + SS * BB * 3 * HH;      // S*B*H  = 98304
    float* h1   = h0 + SS * BB * HH;
    float* h2   = h1 + SS * BB * HH;
    float* Xc   = h2 + SS * BB * HH;          // B*T*S  = 98304
    float* wcol = Xc + BB * TT * SS;          // 512
    float* A3   = wcol + HH;                  // 576
    int*   cur  = (int*)(A3 + SS * BB * SS);  // 512 ints

    dim3 gGemm(3 * HH / 16, BB / 16, SS);     // (96, 4, 3)

    // Layer 0: gi = x0 @ Wih0^T + bih0  (A rows stride T*I picks x[:,0,:])
    wmma_gemm_bias<<<gGemm, 32, 0, stream>>>(x, 0L, TT * II,
                                             Wih0, (long)(3 * HH) * II, II,
                                             bih0, 3 * HH,
                                             gi, (long)BB * 3 * HH, 3 * HH);
    gru_kernel<<<SS * BB * HH / 256, 256, 0, stream>>>(gi, bhh0, 3 * HH, h0);

    // Layer 1
    wmma_gemm_bias<<<gGemm, 32, 0, stream>>>(h0, (long)BB * HH, HH,
                                             WihL, (long)(LL - 1) * 3 * HH * HH, HH,
                                             bihL, (LL - 1) * 3 * HH,
                                             gi, (long)BB * 3 * HH, 3 * HH);
    gru_kernel<<<SS * BB * HH / 256, 256, 0, stream>>>(gi, bhhL, (LL - 1) * 3 * HH, h1);

    // Layer 2
    wmma_gemm_bias<<<gGemm, 32, 0, stream>>>(h1, (long)BB * HH, HH,
                                             WihL + (size_t)3 * HH * HH,
                                             (long)(LL - 1) * 3 * HH * HH, HH,
                                             bihL + 3 * HH, (LL - 1) * 3 * HH,
                                             gi, (long)BB * 3 * HH, 3 * HH);
    gru_kernel<<<SS * BB * HH / 256, 256, 0, stream>>>(gi, bhhL + 3 * HH,
                                                       (LL - 1) * 3 * HH, h2);

    // Scan precomputation
    colsum_kernel<<<2, 256, 0, stream>>>(Wl, wcol);
    xc_gemm<<<BB * TT / 16, 32, 0, stream>>>(x, Ws, Xc);
    attn_sh_kernel<<<SS * BB, 256, 0, stream>>>(h0, h1, h2, wcol, Ws, bs, A3);

    // Sequential selection (tiny) then bulk gather-write
    scan_kernel<<<1, 64, 0, stream>>>(A3, Xc, cur);
    size_t total = (size_t)BB * TT * HH + (size_t)BB * LL * HH;
    int nb = (int)((total + 255) / 256);
    write_out<<<nb, 256, 0, stream>>>(h0, h1, h2, cur, out);
}